// SwitchRouter_43078521979510
// MI455X (gfx1250) — compile-verified
//
#include <hip/hip_runtime.h>
#include <cfloat>

#define DIM          2048
#define NEXP         64
#define TPB          128      // 4 waves of 32
#define TOK_PER_BLK  64       // 16 tokens per wave
#define KSTEP        16       // physical k per macro-step (4 WMMAs)

typedef float v2f __attribute__((ext_vector_type(2)));
typedef float v4f __attribute__((ext_vector_type(4)));
typedef float v8f __attribute__((ext_vector_type(8)));

__device__ __forceinline__ v8f wmma_f32(v8f c, v2f a, v2f b) {
    // (neg_a, A, neg_b, B, c_mod, C, reuse_a, reuse_b)
    return __builtin_amdgcn_wmma_f32_16x16x4_f32(false, a, false, b,
                                                 (short)0, c, false, false);
}

__global__ __launch_bounds__(TPB)
void SwitchRouter_43078521979510_kernel(const float* __restrict__ x,
                                        const float* __restrict__ W,
                                        const float* __restrict__ bias,
                                        int*   __restrict__ out_idx,
                                        float* __restrict__ out_val)
{
    // logits staged as [expert][local_token] : 64*64*4B = 16 KB
    __shared__ float sm[NEXP * TOK_PER_BLK];

    const int tid  = threadIdx.x;
    const int wave = tid >> 5;
    const int lane = tid & 31;
    const int n    = lane & 15;   // A: token row within wave tile; B/C: expert col
    const int hi   = lane >> 4;

    const int tok0  = blockIdx.x * TOK_PER_BLK;
    const int twave = tok0 + wave * 16;

    // Lane-private streaming pointers (8*hi = contiguous-load k-permutation)
    const float* xrow = x + (size_t)(twave + n) * DIM + hi * 8;
    const float* wr0  = W + (size_t)( 0 + n) * DIM + hi * 8;
    const float* wr1  = W + (size_t)(16 + n) * DIM + hi * 8;
    const float* wr2  = W + (size_t)(32 + n) * DIM + hi * 8;
    const float* wr3  = W + (size_t)(48 + n) * DIM + hi * 8;

    v8f acc0 = {}, acc1 = {}, acc2 = {}, acc3 = {};

    #pragma unroll 1
    for (int kb = 0; kb < DIM; kb += KSTEP) {
        v4f A0 = *(const v4f*)(xrow + kb);
        v4f A1 = *(const v4f*)(xrow + kb + 4);
        v4f B00 = *(const v4f*)(wr0 + kb), B01 = *(const v4f*)(wr0 + kb + 4);
        v4f B10 = *(const v4f*)(wr1 + kb), B11 = *(const v4f*)(wr1 + kb + 4);
        v4f B20 = *(const v4f*)(wr2 + kb), B21 = *(const v4f*)(wr2 + kb + 4);
        v4f B30 = *(const v4f*)(wr3 + kb), B31 = *(const v4f*)(wr3 + kb + 4);

        float fa[8] = {A0.x, A0.y, A0.z, A0.w, A1.x, A1.y, A1.z, A1.w};
        float g0[8] = {B00.x, B00.y, B00.z, B00.w, B01.x, B01.y, B01.z, B01.w};
        float g1[8] = {B10.x, B10.y, B10.z, B10.w, B11.x, B11.y, B11.z, B11.w};
        float g2[8] = {B20.x, B20.y, B20.z, B20.w, B21.x, B21.y, B21.z, B21.w};
        float g3[8] = {B30.x, B30.y, B30.z, B30.w, B31.x, B31.y, B31.z, B31.w};

        #pragma unroll
        for (int j = 0; j < 4; ++j) {
            v2f a = {fa[2*j], fa[2*j + 1]};
            acc0 = wmma_f32(acc0, a, (v2f){g0[2*j], g0[2*j + 1]});
            acc1 = wmma_f32(acc1, a, (v2f){g1[2*j], g1[2*j + 1]});
            acc2 = wmma_f32(acc2, a, (v2f){g2[2*j], g2[2*j + 1]});
            acc3 = wmma_f32(acc3, a, (v2f){g3[2*j], g3[2*j + 1]});
        }
    }

    // Bias: in the C layout, expert index (= n + 16t) is constant per lane.
    const float b0 = bias[ 0 + n], b1 = bias[16 + n],
                b2 = bias[32 + n], b3 = bias[48 + n];
    #pragma unroll
    for (int r = 0; r < 8; ++r) {
        acc0[r] += b0; acc1[r] += b1; acc2[r] += b2; acc3[r] += b3;
    }

    // C rows r=0..7 are consecutive local tokens (wave*16 + 8*hi + r):
    // store each tile as two contiguous b128 DS stores at sm[expert][token].
    const int ltbase = wave * 16 + 8 * hi;
    {
        float* d;
        d = &sm[( 0 + n) * TOK_PER_BLK + ltbase];
        *(v4f*)d       = (v4f){acc0[0], acc0[1], acc0[2], acc0[3]};
        *(v4f*)(d + 4) = (v4f){acc0[4], acc0[5], acc0[6], acc0[7]};
        d = &sm[(16 + n) * TOK_PER_BLK + ltbase];
        *(v4f*)d       = (v4f){acc1[0], acc1[1], acc1[2], acc1[3]};
        *(v4f*)(d + 4) = (v4f){acc1[4], acc1[5], acc1[6], acc1[7]};
        d = &sm[(32 + n) * TOK_PER_BLK + ltbase];
        *(v4f*)d       = (v4f){acc2[0], acc2[1], acc2[2], acc2[3]};
        *(v4f*)(d + 4) = (v4f){acc2[4], acc2[5], acc2[6], acc2[7]};
        d = &sm[(48 + n) * TOK_PER_BLK + ltbase];
        *(v4f*)d       = (v4f){acc3[0], acc3[1], acc3[2], acc3[3]};
        *(v4f*)(d + 4) = (v4f){acc3[4], acc3[5], acc3[6], acc3[7]};
    }

    __syncthreads();

    // Phase 2: one thread per token — top-2 + softmax over 64 experts.
    if (tid < TOK_PER_BLK) {
        const int lt = tid;
        float m1 = -FLT_MAX, m2 = -FLT_MAX;
        int   i1 = 0, i2 = 0;
        #pragma unroll 4
        for (int e = 0; e < NEXP; ++e) {
            float v = sm[e * TOK_PER_BLK + lt];   // conflict-free: addr = e*64 + lane
            if (v > m1)      { m2 = m1; i2 = i1; m1 = v; i1 = e; }
            else if (v > m2) { m2 = v;  i2 = e; }
        }
        float s = 0.0f;
        #pragma unroll 4
        for (int e = 0; e < NEXP; ++e)
            s += __expf(sm[e * TOK_PER_BLK + lt] - m1);
        const float inv = 1.0f / s;
        const int tok = tok0 + lt;
        out_idx[2 * tok]     = i1;
        out_idx[2 * tok + 1] = i2;
        out_val[2 * tok]     = inv;                  // exp(m1-m1)/s
        out_val[2 * tok + 1] = __expf(m2 - m1) * inv;
    }
}

extern "C" void kernel_launch(void* const* d_in, const int* in_sizes, int n_in,
                              void* d_out, int out_size, void* d_ws, size_t ws_size,
                              hipStream_t stream) {
    const float* x    = (const float*)d_in[0];
    const float* W    = (const float*)d_in[1];
    const float* bias = (const float*)d_in[2];

    const int T = in_sizes[0] / DIM;          // 16384 tokens (divisible by 64)

    int*   out_idx = (int*)d_out;             // first T*2 elements: int32 indices
    float* out_val = (float*)d_out + (size_t)T * 2;  // next T*2: f32 values

    const int blocks = T / TOK_PER_BLK;
    SwitchRouter_43078521979510_kernel<<<blocks, TPB, 0, stream>>>(
        x, W, bias, out_idx, out_val);
}